// SparseMoELayer_4440996184652
// MI455X (gfx1250) — compile-verified
//
#include <hip/hip_runtime.h>
#include <math.h>

#define D_MODELc 1024
#define D_FFc    2048
#define N_EXP    8
#define TOPK     2
#define NTOK     4096   // B*T = 2*2048

typedef __attribute__((ext_vector_type(2))) float v2f;
typedef __attribute__((ext_vector_type(8))) float v8f;

#define MTILE     32     // tokens per workgroup (2 x 16-row WMMA subtiles)
// Subtile rows interleaved: logical row r -> physical row 2*(r%16) + r/16.
// XS phys-row pitch 2*1026 == 4 (mod 64) -> 64b fragment loads hit all 64 banks;
// subtile-1 fragment sits +4104B away -> folds into the DS offset field.
#define XS_STRIDE 1026
#define KCHUNK    128    // d_ff chunk held in LDS
#define HS_STRIDE 130    // 2*130 == 4 (mod 64), +520B for subtile 1

// ---------------- zero out + counts ----------------
__global__ void k_zero(float* __restrict__ out, int* __restrict__ counts) {
  int i = blockIdx.x * 256 + threadIdx.x;
  if (i < NTOK * D_MODELc) out[i] = 0.0f;
  if (i < N_EXP) counts[i] = 0;
}

// ---------------- router: one wave per token ----------------
__global__ void k_router(const float* __restrict__ x, const float* __restrict__ gw,
                         float* __restrict__ probs, int* __restrict__ topi,
                         float* __restrict__ topv, int* __restrict__ counts) {
  int wid = threadIdx.x >> 5, lane = threadIdx.x & 31;
  int t = blockIdx.x * 8 + wid;
  if (t >= NTOK) return;
  const float* xr = x + (size_t)t * D_MODELc;
  float part[N_EXP];
#pragma unroll
  for (int e = 0; e < N_EXP; ++e) part[e] = 0.0f;
  for (int i = 0; i < D_MODELc / 32; ++i) {
    int d = lane + 32 * i;
    float xv = xr[d];
#pragma unroll
    for (int e = 0; e < N_EXP; ++e) part[e] += xv * gw[e * D_MODELc + d];
  }
#pragma unroll
  for (int e = 0; e < N_EXP; ++e) {
    float v = part[e];
    for (int off = 16; off > 0; off >>= 1) v += __shfl_xor(v, off, 32);
    part[e] = v;
  }
  if (lane == 0) {
    float m = part[0];
#pragma unroll
    for (int e = 1; e < N_EXP; ++e) m = fmaxf(m, part[e]);
    float p[N_EXP], s = 0.0f;
#pragma unroll
    for (int e = 0; e < N_EXP; ++e) { p[e] = __expf(part[e] - m); s += p[e]; }
    float inv = 1.0f / s;
#pragma unroll
    for (int e = 0; e < N_EXP; ++e) { p[e] *= inv; probs[t * N_EXP + e] = p[e]; }
    int i1 = 0; float v1 = p[0];
#pragma unroll
    for (int e = 1; e < N_EXP; ++e) if (p[e] > v1) { v1 = p[e]; i1 = e; }
    int i2 = -1; float v2 = -1.0f;
#pragma unroll
    for (int e = 0; e < N_EXP; ++e)
      if (e != i1 && p[e] > v2) { v2 = p[e]; i2 = e; }
    topi[t * 2 + 0] = i1; topi[t * 2 + 1] = i2;
    topv[t * 2 + 0] = v1; topv[t * 2 + 1] = v2;
    atomicAdd(&counts[i1], 1);
    atomicAdd(&counts[i2], 1);
  }
}

// ---------------- offsets + deterministic router loss ----------------
__global__ void k_setup(const float* __restrict__ probs, const int* __restrict__ counts,
                        int* __restrict__ offsets, int* __restrict__ cursors,
                        float* __restrict__ out_loss) {
  __shared__ float ps[N_EXP];
  int tid = threadIdx.x;
  if (tid < N_EXP) {
    float s = 0.0f;
    for (int t = 0; t < NTOK; ++t) s += probs[t * N_EXP + tid];
    ps[tid] = s / (float)NTOK;
  }
  __syncthreads();
  if (tid == 0) {
    int run = 0;
    float loss = 0.0f;
    for (int e = 0; e < N_EXP; ++e) {
      offsets[e] = run; cursors[e] = run; run += counts[e];
      float f = (float)counts[e] / (float)(NTOK * TOPK);
      loss += f * ps[e];   // E * mean_e(f*p) == sum_e(f*p)
    }
    *out_loss = loss;
  }
}

// ---------------- bucket scatter ----------------
__global__ void k_scatter(const int* __restrict__ topi, const float* __restrict__ topv,
                          int* __restrict__ cursors, int* __restrict__ btok,
                          float* __restrict__ bcw) {
  int t = blockIdx.x * 256 + threadIdx.x;
  if (t >= NTOK) return;
#pragma unroll
  for (int k = 0; k < TOPK; ++k) {
    int e = topi[t * 2 + k];
    int pos = atomicAdd(&cursors[e], 1);
    btok[pos] = t;
    bcw[pos] = topv[t * 2 + k];
  }
}

// ---- fused expert FFN: silu(X@W1)@W2, WMMA f32 16x16x4, M=32 blocking ----
__global__ void __launch_bounds__(256, 1)
k_ffn(const float* __restrict__ x, const float* __restrict__ w1,
      const float* __restrict__ w2, const int* __restrict__ counts,
      const int* __restrict__ offsets, const int* __restrict__ btok,
      const float* __restrict__ bcw, float* __restrict__ out) {
  __shared__ float Xs[MTILE * XS_STRIDE];   // 32 x 1024 X tile, rows interleaved
  __shared__ float Hs[MTILE * HS_STRIDE];   // 32 x 128 H chunk, rows interleaved
  __shared__ int   tk[MTILE];
  __shared__ float cwv[MTILE];

  const int e = blockIdx.y;
  const int tile = blockIdx.x;
  const int cnt = counts[e];
  if (tile * MTILE >= cnt) return;
  const int base = offsets[e] + tile * MTILE;

  const int tid = threadIdx.x;
  const int wid = tid >> 5, lane = tid & 31;
  const int half = lane >> 4, ln = lane & 15;

  if (tid < MTILE) {
    int r = tile * MTILE + tid;
    if (r < cnt) { tk[tid] = btok[base + tid]; cwv[tid] = bcw[base + tid]; }
    else         { tk[tid] = 0;               cwv[tid] = 0.0f; }
  }
  __syncthreads();

  // stage 32 x 1024 X tile into LDS (float2, 8 threads per physical row)
  {
    int prow = tid >> 3, c8 = tid & 7;
    int lrow = (prow >> 1) + (prow & 1) * 16;      // physical -> logical row
    const float2* src = (const float2*)(x + (size_t)tk[lrow] * D_MODELc);
    float* dst = &Xs[prow * XS_STRIDE];
#pragma unroll
    for (int i = 0; i < 64; ++i) {
      int c2 = c8 + 8 * i;
      *(float2*)&dst[c2 * 2] = src[c2];
    }
  }
  __syncthreads();

  const float* W1e = w1 + (size_t)e * D_MODELc * D_FFc;
  const float* W2e = w2 + (size_t)e * D_FFc * D_MODELc;

  const v8f vzero = {};
  v8f yacc[16];                       // [out-tile 0..7][subtile 0..1]
#pragma unroll
  for (int i = 0; i < 16; ++i) yacc[i] = vzero;

  for (int kc = 0; kc < D_FFc / KCHUNK; ++kc) {
    // ---- phase 1: H[32][128] = silu(X @ W1[:, chunk]); one 16-col tile per wave ----
    {
      int ncol = kc * KCHUNK + wid * 16 + ln;         // global d_ff column
      const float* bcol = W1e + ncol;
      const float* arow = &Xs[(2 * ln) * XS_STRIDE];  // subtile 1 = +XS_STRIDE
      v8f acc0 = vzero, acc1 = vzero;
      for (int k = 0; k < D_MODELc; k += 4) {
        int kr = k + 2 * half;
        v2f b;
        b.x = bcol[(size_t)kr * D_FFc];
        b.y = bcol[(size_t)(kr + 1) * D_FFc];
        v2f a0 = *(const v2f*)&arow[kr];
        v2f a1 = *(const v2f*)&arow[XS_STRIDE + kr];
        acc0 = __builtin_amdgcn_wmma_f32_16x16x4_f32(false, a0, false, b,
                                                     (short)0, acc0, false, false);
        acc1 = __builtin_amdgcn_wmma_f32_16x16x4_f32(false, a1, false, b,
                                                     (short)0, acc1, false, false);
      }
#pragma unroll
      for (int j = 0; j < 8; ++j) {
        float v0 = acc0[j];
        float v1 = acc1[j];
        v0 = v0 * __builtin_amdgcn_rcpf(1.0f + __expf(-v0));   // SiLU (native rcp)
        v1 = v1 * __builtin_amdgcn_rcpf(1.0f + __expf(-v1));
        // logical row j+8*half (sub0) / +16 (sub1) -> physical 2*(j+8*half)+sub
        Hs[(2 * (j + 8 * half) + 0) * HS_STRIDE + wid * 16 + ln] = v0;
        Hs[(2 * (j + 8 * half) + 1) * HS_STRIDE + wid * 16 + ln] = v1;
      }
    }
    __syncthreads();

    // ---- phase 2: Y += H_chunk @ W2[chunk, :]; 8 out-tiles x 2 subtiles per wave ----
#pragma unroll
    for (int ot = 0; ot < 8; ++ot) {
      int ncol = (wid * 8 + ot) * 16 + ln;            // 0..1023 d_model column
      const float* bcol = W2e + (size_t)kc * KCHUNK * D_MODELc + ncol;
      const float* arow = &Hs[(2 * ln) * HS_STRIDE];
      v8f acc0 = yacc[ot * 2 + 0];
      v8f acc1 = yacc[ot * 2 + 1];
      for (int k = 0; k < KCHUNK; k += 4) {
        int kr = k + 2 * half;
        v2f b;
        b.x = bcol[(size_t)kr * D_MODELc];
        b.y = bcol[(size_t)(kr + 1) * D_MODELc];
        v2f a0 = *(const v2f*)&arow[kr];
        v2f a1 = *(const v2f*)&arow[HS_STRIDE + kr];
        acc0 = __builtin_amdgcn_wmma_f32_16x16x4_f32(false, a0, false, b,
                                                     (short)0, acc0, false, false);
        acc1 = __builtin_amdgcn_wmma_f32_16x16x4_f32(false, a1, false, b,
                                                     (short)0, acc1, false, false);
      }
      yacc[ot * 2 + 0] = acc0;
      yacc[ot * 2 + 1] = acc1;
    }
    __syncthreads();
  }

  // ---- scaled scatter-add into output (2 commutative adds/element -> deterministic) ----
#pragma unroll
  for (int ot = 0; ot < 8; ++ot) {
    int ncol = (wid * 8 + ot) * 16 + ln;
#pragma unroll
    for (int sub = 0; sub < 2; ++sub) {
#pragma unroll
      for (int j = 0; j < 8; ++j) {
        int row = j + 8 * half + 16 * sub;            // logical row
        float v = yacc[ot * 2 + sub][j] * cwv[row];
        unsafeAtomicAdd(&out[(size_t)tk[row] * D_MODELc + ncol], v);
      }
    }
  }
}

extern "C" void kernel_launch(void* const* d_in, const int* in_sizes, int n_in,
                              void* d_out, int out_size, void* d_ws, size_t ws_size,
                              hipStream_t stream) {
  const float* x  = (const float*)d_in[0];
  const float* gw = (const float*)d_in[1];
  const float* w1 = (const float*)d_in[2];
  const float* w2 = (const float*)d_in[3];
  float* out = (float*)d_out;

  // workspace layout (4-byte units)
  float* ws     = (float*)d_ws;
  float* probs  = ws;                              // N*8 f32
  int*   topi   = (int*)(ws + NTOK * 8);           // N*2 i32
  float* topv   = ws + NTOK * 8 + NTOK * 2;        // N*2 f32
  int*   counts = (int*)(ws + NTOK * 8 + NTOK * 4);// 8
  int*   offs   = counts + 8;                      // 8
  int*   curs   = counts + 16;                     // 8 (+pad)
  int*   btok   = counts + 32;                     // N*2 i32
  float* bcw    = (float*)(btok + NTOK * 2);       // N*2 f32

  k_zero   <<<(NTOK * D_MODELc + 255) / 256, 256, 0, stream>>>(out, counts);
  k_router <<<NTOK / 8, 256, 0, stream>>>(x, gw, probs, topi, topv, counts);
  k_setup  <<<1, 256, 0, stream>>>(probs, counts, offs, curs,
                                   out + (size_t)NTOK * D_MODELc);
  k_scatter<<<NTOK / 256, 256, 0, stream>>>(topi, topv, curs, btok, bcw);
  dim3 g(128, N_EXP);  // 128 tiles of 32 covers worst case count=4096 per expert
  k_ffn    <<<g, 256, 0, stream>>>(x, w1, w2, counts, offs, btok, bcw, out);
}